// CollaborativeRNNModel_2834678415600
// MI455X (gfx1250) — compile-verified
//
#include <hip/hip_runtime.h>
#include <hip/hip_bf16.h>

typedef __attribute__((ext_vector_type(2))) float v2f;
typedef __attribute__((ext_vector_type(8))) float v8f;

#define HH 128      // hidden units
#define H2 256      // 2*H
#define BB 64       // batch
#define TT 100      // seq len
#define NI 20001    // num items + 1
#define LDA 132     // padded LDS row stride (stride mod 64-bank period = 4 -> conflict free)

// ---------------------------------------------------------------------------
// Kernel 1: recurrent scan. One workgroup per batch row (rows independent,
// only time is sequential). 256 threads: waves 0-3 compute the 128 live
// update-gate outputs (reset gate is dead code in the reference), waves 4-7
// compute the 128 candidate outputs. Gathered per-user kernels are read with
// non-temporal hints (random uids, ~no reuse; don't thrash L2).
// ---------------------------------------------------------------------------
__global__ __launch_bounds__(256) void crnn_recurrent(
    const int* __restrict__ user_ids, const int* __restrict__ item_ids,
    const float* __restrict__ h0,
    const float* __restrict__ gate_ku, const float* __restrict__ gate_ki,
    const float* __restrict__ gate_bias,
    const float* __restrict__ cand_ku, const float* __restrict__ cand_ki,
    const float* __restrict__ cand_bias,
    float* __restrict__ states)        // [B*T, H]
{
    __shared__ float h_sh[HH];
    __shared__ float u_sh[HH];
    __shared__ float c_sh[HH];

    const int b = blockIdx.x;
    const int k = threadIdx.x;

    if (k < HH) h_sh[k] = h0[b * HH + k];
    __syncthreads();

    for (int t = 0; t < TT; ++t) {
        const int uid = user_ids[b * TT + t];
        const int iid = item_ids[b * TT + t];

        if (k < HH) {
            // update gate u_k = sigmoid( h . gate_ku[uid][:, H+k] + bias + ki )
            const float* kp = gate_ku + ((size_t)uid * HH) * H2 + (HH + k);
            float acc = gate_bias[HH + k] + gate_ki[(size_t)iid * H2 + HH + k];
            #pragma unroll 8
            for (int h = 0; h < HH; ++h)
                acc += h_sh[h] * __builtin_nontemporal_load(&kp[(size_t)h * H2]);
            u_sh[k] = 1.0f / (1.0f + __expf(-acc));
        } else {
            const int j = k - HH;
            const float* kp = cand_ku + ((size_t)uid * HH) * HH + j;
            float acc = cand_bias[j] + cand_ki[(size_t)iid * HH + j];
            #pragma unroll 8
            for (int h = 0; h < HH; ++h)
                acc += h_sh[h] * __builtin_nontemporal_load(&kp[(size_t)h * HH]);
            c_sh[j] = tanhf(acc);
        }
        __syncthreads();

        if (k < HH) {
            const float u  = u_sh[k];
            const float hn = u * h_sh[k] + (1.0f - u) * c_sh[k];
            h_sh[k] = hn;
            // scan states -> [B,T,H] row-major == reference row index b*T+t
            states[((size_t)b * TT + t) * HH + k] = hn;
        }
        __syncthreads();
    }
}

// ---------------------------------------------------------------------------
// Kernel 2: logits = states [6400,128] @ ws [128,20001], fp32 via
// V_WMMA_F32_16X16X4_F32 chained over K (32 steps). Block = 256 threads
// (8 waves). Each wave owns TWO adjacent 16x16 output tiles that share one
// A fragment (1 ds_load : 2 wmma per K-step). blockIdx.x -> M tile (A staged
// once into padded LDS), blockIdx.y -> 16 N tiles per block.
//
// Bounds handling: WMMA output column j depends only on B column j (lanes j,
// j+16), so out-of-range lanes load a CLAMPED valid address (no EXEC masking
// in the hot loop) and their garbage columns are simply never stored.
// Output (512 MB, write-once) is streamed with non-temporal stores.
// ---------------------------------------------------------------------------
__global__ __launch_bounds__(256) void crnn_logits(
    const float* __restrict__ X,    // [6400, 128]
    const float* __restrict__ Wm,   // [128, NI]
    float* __restrict__ out)        // [6400, NI]
{
    __shared__ float Ash[16 * LDA];

    const int tid  = threadIdx.x;
    const int row0 = blockIdx.x * 16;

    // A tile is 16 consecutive full rows of X -> one contiguous 8KB block
    const float* src = X + (size_t)row0 * HH;
    for (int i = tid; i < 16 * HH; i += 256)
        Ash[(i >> 7) * LDA + (i & 127)] = src[i];
    __syncthreads();

    const int wave = tid >> 5;
    const int lane = tid & 31;
    const int col0 = (blockIdx.y * 16 + wave * 2) * 16;  // first of two tiles
    if (col0 >= NI) return;                              // wave-uniform

    const int m    = lane & 15;             // A row / B col within tile
    const int half = lane >> 4;             // lanes 16-31 carry K+2,K+3
    const int koff = half * 2;
    const int n0   = col0 + m;
    const int n1   = col0 + 16 + m;
    const int nl0  = (n0 < NI) ? n0 : (NI - 1);   // clamped load columns
    const int nl1  = (n1 < NI) ? n1 : (NI - 1);

    v8f acc0 = {};
    v8f acc1 = {};
    const float* arow = Ash + m * LDA + koff;

    #pragma unroll 8
    for (int k4 = 0; k4 < HH; k4 += 4) {
        // A fragment: lane m holds X[row0+m][k4+koff .. +1] (8B aligned LDS read)
        v2f a = *(const v2f*)(arow + k4);
        // B fragments: lane holds ws[k4+koff][n], ws[k4+koff+1][n]
        const size_t base = (size_t)(k4 + koff) * NI;
        v2f b0, b1;
        b0.x = Wm[base + nl0];
        b0.y = Wm[base + NI + nl0];
        b1.x = Wm[base + nl1];
        b1.y = Wm[base + NI + nl1];
        acc0 = __builtin_amdgcn_wmma_f32_16x16x4_f32(
                   false, a, false, b0, (short)0, acc0, false, false);
        acc1 = __builtin_amdgcn_wmma_f32_16x16x4_f32(
                   false, a, false, b1, (short)0, acc1, false, false);
    }

    // D layout: VGPR r -> row (r + 8*half), col n
    const int r0 = row0 + half * 8;
    if (n0 < NI) {
        #pragma unroll
        for (int r = 0; r < 8; ++r)
            __builtin_nontemporal_store(acc0[r], &out[(size_t)(r0 + r) * NI + n0]);
    }
    if (n1 < NI) {
        #pragma unroll
        for (int r = 0; r < 8; ++r)
            __builtin_nontemporal_store(acc1[r], &out[(size_t)(r0 + r) * NI + n1]);
    }
}

// ---------------------------------------------------------------------------
extern "C" void kernel_launch(void* const* d_in, const int* in_sizes, int n_in,
                              void* d_out, int out_size, void* d_ws, size_t ws_size,
                              hipStream_t stream) {
    const int*   user_ids  = (const int*)  d_in[0];
    const int*   item_ids  = (const int*)  d_in[1];
    const float* h0        = (const float*)d_in[2];
    const float* gate_ku   = (const float*)d_in[3];
    const float* gate_ki   = (const float*)d_in[4];
    const float* gate_bias = (const float*)d_in[5];
    const float* cand_ku   = (const float*)d_in[6];
    const float* cand_ki   = (const float*)d_in[7];
    const float* cand_bias = (const float*)d_in[8];
    const float* wsm       = (const float*)d_in[9];

    float* out    = (float*)d_out;
    float* states = (float*)d_ws;   // needs B*T*H*4 = 3.3 MB scratch

    crnn_recurrent<<<BB, 256, 0, stream>>>(
        user_ids, item_ids, h0, gate_ku, gate_ki, gate_bias,
        cand_ku, cand_ki, cand_bias, states);

    // 8 waves x 2 tiles x 16 cols = 256 columns per block
    dim3 g2((BB * TT) / 16, (NI + 255) / 256);   // 400 x 79
    crnn_logits<<<g2, 256, 0, stream>>>(states, wsm, out);
}